// VarCorAttention_23742579212905
// MI455X (gfx1250) — compile-verified
//
#include <hip/hip_runtime.h>

// VarCorAttention on MI455X (gfx1250, wave32).
// Math reduction: mean(irfft(...)) keeps only the DC term, so
//   corr[b,i,j] = (para0 / T) * para2[i,j] * rowsum(Q)[b,i] * rowsum(K)[b,j]
//   attn = softmax(corr / sqrt(T)); out[b] = attn[b] @ values[b].
// Memory-bound: 64 MB total traffic -> ~2.7 us at 23.3 TB/s HBM.
// GEMM uses V_WMMA_F32_16X16X4_F32 (full fp32 precision; compute is free).

#define B_DIM 16
#define D_DIM 128
#define T_DIM 512
#define LDS_STRIDE 132  // 128 + 4 pad: row stride hits distinct LDS banks

typedef __attribute__((ext_vector_type(2))) float v2f;
typedef __attribute__((ext_vector_type(8))) float v8f;

// ---------------------------------------------------------------------------
// Kernel 1: row sums of Q and K along T. One block per (b,d) row.
// 128 threads x float4 = 512 floats per row. Fully coalesced streaming pass.
// ---------------------------------------------------------------------------
__global__ __launch_bounds__(128)
void varcor_rowsum_kernel(const float* __restrict__ q,
                          const float* __restrict__ k,
                          float* __restrict__ sq,
                          float* __restrict__ sk) {
    __shared__ float redq[128];
    __shared__ float redk[128];
    const int row = blockIdx.x;          // b*D + d  (0..2047)
    const int tid = threadIdx.x;         // 0..127
    const float4 a = ((const float4*)(q + (size_t)row * T_DIM))[tid];
    const float4 b = ((const float4*)(k + (size_t)row * T_DIM))[tid];
    redq[tid] = a.x + a.y + a.z + a.w;
    redk[tid] = b.x + b.y + b.z + b.w;
    __syncthreads();
    for (int s = 64; s > 0; s >>= 1) {
        if (tid < s) {
            redq[tid] += redq[tid + s];
            redk[tid] += redk[tid + s];
        }
        __syncthreads();
    }
    if (tid == 0) {
        sq[row] = redq[0];
        sk[row] = redk[0];
    }
}

// ---------------------------------------------------------------------------
// Kernel 2: logits + softmax over j. One block (128 threads) per (b,i) row.
// ---------------------------------------------------------------------------
__global__ __launch_bounds__(128)
void varcor_softmax_kernel(const float* __restrict__ sq,
                           const float* __restrict__ sk,
                           const float* __restrict__ para,
                           const float* __restrict__ para2,
                           float* __restrict__ attn) {
    __shared__ float red[128];
    const int row = blockIdx.x;          // b*D + i
    const int b   = row >> 7;
    const int i   = row & (D_DIM - 1);
    const int j   = threadIdx.x;

    // coef = para0 / T (DC mean) * 1/sqrt(T) (attention scale)
    const float inv_t_sqrt_t = 1.0f / (512.0f * 22.62741699796952f);
    const float coef = para[0] * inv_t_sqrt_t;

    float x = coef * para2[i * D_DIM + j] * sq[row] * sk[b * D_DIM + j];

    red[j] = x;
    __syncthreads();
    for (int s = 64; s > 0; s >>= 1) {
        if (j < s) red[j] = fmaxf(red[j], red[j + s]);
        __syncthreads();
    }
    const float m = red[0];
    __syncthreads();

    const float e = __expf(x - m);
    red[j] = e;
    __syncthreads();
    for (int s = 64; s > 0; s >>= 1) {
        if (j < s) red[j] += red[j + s];
        __syncthreads();
    }
    attn[(size_t)row * D_DIM + j] = e / red[0];
}

// ---------------------------------------------------------------------------
// Kernel 3: out[b] = attn[b] (128x128) @ values[b] (128x512) via
// V_WMMA_F32_16X16X4_F32. Block = 128 threads = 4 waves; block computes a
// 16(M) x 64(N) output tile, each wave one 16x16 tile, K unrolled by 4 per
// WMMA. A tile staged in LDS (padded stride), B streamed coalesced from HBM.
//
// Fragment layouts (ISA 7.12.2, 32-bit, wave32):
//   A 16x4 : lane L -> M = L%16; VGPR v holds K = v + 2*(L/16)
//   B 4x16 : lane L -> N = L%16; VGPR v holds K = v + 2*(L/16)
//   C 16x16: lane L -> N = L%16; VGPR r holds M = r + 8*(L/16)
// ---------------------------------------------------------------------------
__global__ __launch_bounds__(128)
void varcor_wmma_av_kernel(const float* __restrict__ attn,
                           const float* __restrict__ values,
                           float* __restrict__ out) {
    __shared__ float lA[16 * LDS_STRIDE];   // 16 rows of A, padded

    const int tid  = threadIdx.x;           // 0..127
    const int wave = tid >> 5;              // 0..3
    const int lane = tid & 31;

    const int blk   = blockIdx.x;           // B * (D/16) * (T/64) = 1024
    const int b     = blk >> 6;
    const int rem   = blk & 63;
    const int mtile = rem >> 3;             // 0..7
    const int ntile = rem & 7;              // 0..7
    const int m0    = mtile * 16;
    const int n0    = ntile * 64 + wave * 16;

    // Cooperative, coalesced load of the 16x128 A tile into LDS.
    const float* Ab = attn + (size_t)b * D_DIM * D_DIM;
    #pragma unroll
    for (int qq = 0; qq < 4; ++qq) {
        const int flat = qq * 512 + tid * 4;       // 0..2047
        const int r    = flat >> 7;                // /128
        const int c    = flat & (D_DIM - 1);
        const float4 v = *(const float4*)(Ab + (size_t)(m0 + r) * D_DIM + c);
        *(float4*)(&lA[r * LDS_STRIDE + c]) = v;
    }
    __syncthreads();

    const int mn   = lane & 15;             // M for A-frag, N for B/C-frags
    const int half = lane >> 4;             // 0 or 1

    const float* Vb = values + (size_t)b * D_DIM * T_DIM + n0 + mn;

    v8f acc = {};
    #pragma unroll 4
    for (int k = 0; k < D_DIM; k += 4) {
        const int kk = k + 2 * half;
        // 8-byte aligned LDS read of the K-pair for this lane half.
        v2f a = *(const v2f*)(&lA[mn * LDS_STRIDE + kk]);
        v2f bb;
        bb.x = Vb[(size_t)(kk + 0) * T_DIM];
        bb.y = Vb[(size_t)(kk + 1) * T_DIM];
        acc = __builtin_amdgcn_wmma_f32_16x16x4_f32(
            /*neg_a=*/false, a, /*neg_b=*/false, bb,
            /*c_mod=*/(short)0, acc, /*reuse_a=*/false, /*reuse_b=*/false);
    }

    float* Ob = out + (size_t)b * D_DIM * T_DIM + n0 + mn;
    #pragma unroll
    for (int r = 0; r < 8; ++r) {
        Ob[(size_t)(m0 + r + 8 * half) * T_DIM] = acc[r];
    }
}

// ---------------------------------------------------------------------------
// Launch. Workspace layout (floats): Sq[2048] | Sk[2048] | attn[16*128*128]
// = ~1.05 MB total.
// ---------------------------------------------------------------------------
extern "C" void kernel_launch(void* const* d_in, const int* in_sizes, int n_in,
                              void* d_out, int out_size, void* d_ws, size_t ws_size,
                              hipStream_t stream) {
    const float* q     = (const float*)d_in[0];   // [16,128,512]
    const float* k     = (const float*)d_in[1];   // [16,128,512]
    const float* v     = (const float*)d_in[2];   // [16,128,512]
    // d_in[3] = attn_mask (unused)
    const float* para  = (const float*)d_in[4];   // [25], only para[0] used
    const float* para2 = (const float*)d_in[5];   // [128,128]
    float*       out   = (float*)d_out;           // [16,128,512]

    float* ws   = (float*)d_ws;
    float* sq   = ws;                              // 2048
    float* sk   = ws + B_DIM * D_DIM;              // 2048
    float* attn = ws + 2 * B_DIM * D_DIM;          // 16*128*128

    varcor_rowsum_kernel<<<B_DIM * D_DIM, 128, 0, stream>>>(q, k, sq, sk);
    varcor_softmax_kernel<<<B_DIM * D_DIM, 128, 0, stream>>>(sq, sk, para, para2, attn);
    varcor_wmma_av_kernel<<<B_DIM * (D_DIM / 16) * (T_DIM / 64), 128, 0, stream>>>(attn, v, out);
}